// DeformableDetrModel_33234456936628
// MI455X (gfx1250) — compile-verified
//
#include <hip/hip_runtime.h>
#include <hip/hip_bf16.h>
#include <math.h>
#include <stdint.h>

// ---------------- problem constants (match reference) ----------------
#define BS     8
#define QN     900
#define NH     8
#define NL     4
#define NPNT   4
#define DH     32
#define TOTAL  19947
#define KDIM   256

typedef __attribute__((ext_vector_type(2))) float v2f;
typedef __attribute__((ext_vector_type(4))) float v4f;
typedef __attribute__((ext_vector_type(8))) float v8f;
typedef __attribute__((ext_vector_type(4))) unsigned int u32x4;
typedef __attribute__((ext_vector_type(8))) unsigned int u32x8;

// Use the CDNA5 Tensor Data Mover for A-tile staging (inline asm path,
// portable across ROCm-7.2 / amdgpu-toolchain per CDNA5_HIP.md).
#define USE_TDM 1

// =====================================================================
// fp32 WMMA GEMM, K fixed at 256:  C[M,N] = A[M,256] @ B[256,N] + bias
// N, store-MODE are compile-time so every B load is base + IOFFSET imm.
// Block: 256 threads = 8 wave32s, covers 16 rows x N cols.
// Each wave owns a 16x32 slab: 2 accumulators, A fragment reused 2x.
// MODE 0: C[row*N+col]
// MODE 1: fused transpose to value layout [bs,NH,TOTAL,DH]
// =====================================================================
template <int N, int MODE>
__global__ __launch_bounds__(256) void gemm256_wmma(
    const float* __restrict__ A, const float* __restrict__ B,
    const float* __restrict__ bias, float* __restrict__ C, int M)
{
    // Row stride 260 floats: TDM pad (4 DWORDs per 256 DWORDs) == this padding.
    __shared__ float As[16][KDIM + 4];

    const int row0 = blockIdx.x * 16;
    const int wave = threadIdx.x >> 5;
    const int lane = threadIdx.x & 31;

#if USE_TDM
    // -------- Tensor Data Mover: stage A tile (16 x 256 f32) into LDS ----
    // D# per CDNA5 ISA. 2D tile, data_size=4B,
    // pad_enable: insert 4 DWORDs every 256 DWORDs (-> LDS row stride 260),
    // tensor_dim1 = remaining rows so the partial last M-tile zero-fills.
    if (threadIdx.x < 32) {   // wave 0 issues the DMA once per block
        const unsigned int lds_off = (unsigned int)(uintptr_t)(&As[0][0]);
        const unsigned long long ga =
            (unsigned long long)(uintptr_t)(A + (size_t)row0 * KDIM);
        int rows = M - row0; rows = rows > 16 ? 16 : rows;

        u32x4 d0;
        d0[0] = 1u;                                   // count=1, user D#
        d0[1] = lds_off;                              // lds_addr (bytes)
        d0[2] = (unsigned int)ga;                     // global_addr[31:0]
        d0[3] = ((unsigned int)(ga >> 32) & 0x01FFFFFFu) | (2u << 30); // [56:32]|type=2

        u32x8 d1;
        d1[0] = (2u << 16)        // data_size = 4 bytes
              | (1u << 20)        // pad_enable
              | (7u << 22)        // pad_interval: every 256 DWORDs
              | (3u << 25);       // pad_amount: 4 DWORDs
        d1[1] = ((unsigned int)KDIM & 0xFFFFu) << 16; // tensor_dim0[15:0]
        d1[2] = ((unsigned int)rows) << 16;           // tensor_dim1[15:0]
        d1[3] = ((unsigned int)KDIM) << 16;           // tile_dim0 = 256
        d1[4] = 16u;                                  // tile_dim1 = 16, tile_dim2 = 0
        d1[5] = (unsigned int)KDIM;                   // tensor_dim0_stride = 256
        d1[6] = 0u;
        d1[7] = 0u;

        asm volatile("tensor_load_to_lds %0, %1" :: "s"(d0), "s"(d1) : "memory");
        __builtin_amdgcn_s_wait_tensorcnt(0);
    }
    __syncthreads();
#else
    // -------- fallback: cooperative b128 staging --------
    for (int i = threadIdx.x * 4; i < 16 * KDIM; i += 256 * 4) {
        const int r = i >> 8;
        const int k = i & 255;
        int gr = row0 + r; gr = (gr < M) ? gr : (M - 1); // clamp: row-local garbage only
        *(v4f*)&As[r][k] = *(const v4f*)&A[(size_t)gr * KDIM + k];
    }
    __syncthreads();
#endif

    const int n0 = wave * 32;
    if (n0 >= N) return;   // wave-uniform exit (N=128 uses 4 of 8 waves)

    // ISA VGPR layouts (CDNA5 ISA §7.12.2):
    //  A 16x4: lanes 0-15 = M rows; lane half selects K pair {0,1}|{2,3}
    //  B 4x16: lanes 0-15 = N cols; lane half selects K pair
    //  C/D:    VGPR r -> M = r | r+8 by lane half, N = lane%16
    const int mrow  = lane & 15;
    const int ncol  = lane & 15;
    const int khalf = (lane >> 4) << 1;

    const float* __restrict__ Bbase = B + (size_t)khalf * N + n0 + ncol;
    const float* __restrict__ Abase = &As[mrow][khalf];
    __builtin_prefetch(Bbase, 0, 3);

    v8f acc0 = {}, acc1 = {};
    #pragma unroll 8
    for (int k0 = 0; k0 < KDIM; k0 += 4) {
        v2f a, b0, b1;
        a.x  = Abase[k0];                        // ds_load_b64, imm offset
        a.y  = Abase[k0 + 1];
        b0.x = Bbase[(size_t)(k0 + 0) * N];      // global_load_b32, imm offsets
        b0.y = Bbase[(size_t)(k0 + 1) * N];
        b1.x = Bbase[(size_t)(k0 + 0) * N + 16];
        b1.y = Bbase[(size_t)(k0 + 1) * N + 16];
        acc0 = __builtin_amdgcn_wmma_f32_16x16x4_f32(
                   false, a, false, b0, (short)0, acc0, false, false);
        acc1 = __builtin_amdgcn_wmma_f32_16x16x4_f32(
                   false, a, false, b1, (short)0, acc1, false, false);
    }

    const int   c0  = n0 + ncol;
    const int   c1  = c0 + 16;
    const float bi0 = bias[c0];
    const float bi1 = bias[c1];
    #pragma unroll
    for (int r = 0; r < 8; ++r) {
        const int row = row0 + ((lane < 16) ? r : (r + 8));
        if (row < M) {
            const float v0 = acc0[r] + bi0;
            const float v1 = acc1[r] + bi1;
            if (MODE == 0) {
                C[(size_t)row * N + c0] = v0;
                C[(size_t)row * N + c1] = v1;
            } else {
                const int b_ = row / TOTAL;
                const int t_ = row - b_ * TOTAL;
                C[(((size_t)(b_ * NH + (c0 >> 5)) * TOTAL) + t_) * DH + (c0 & 31)] = v0;
                C[(((size_t)(b_ * NH + (c1 >> 5)) * TOTAL) + t_) * DH + (c1 & 31)] = v1;
            }
        }
    }
}

// =====================================================================
// Softmax over the 16 (NL*NP) attention logits per (b,q,h).
// =====================================================================
__global__ __launch_bounds__(256) void softmax16_kernel(float* __restrict__ attn, int rows)
{
    const int r = blockIdx.x * blockDim.x + threadIdx.x;
    if (r >= rows) return;
    float* p = attn + (size_t)r * 16;
    float m = p[0];
    #pragma unroll
    for (int i = 1; i < 16; ++i) m = fmaxf(m, p[i]);
    float e[16];
    float s = 0.0f;
    #pragma unroll
    for (int i = 0; i < 16; ++i) { e[i] = __expf(p[i] - m); s += e[i]; }
    const float inv = 1.0f / s;
    #pragma unroll
    for (int i = 0; i < 16; ++i) p[i] = e[i] * inv;
}

// =====================================================================
// Multi-scale deformable bilinear sampling + attention-weighted reduce.
// One wave32 per (b,q,h); lane == channel d (DH == 32 == wave width).
// =====================================================================
__global__ __launch_bounds__(256) void ms_deform_sample_kernel(
    const float* __restrict__ value, const float* __restrict__ refp,
    const float* __restrict__ off,   const float* __restrict__ attn,
    float* __restrict__ mid)
{
    const int Hs[NL] = {100, 50, 25, 13};
    const int Ws[NL] = {150, 75, 38, 19};
    const int St[NL] = {0, 15000, 18750, 19700};

    const int gw   = (int)((blockIdx.x * (size_t)blockDim.x + threadIdx.x) >> 5);
    const int lane = threadIdx.x & 31;

    const int h  = gw & (NH - 1);
    const int bq = gw >> 3;
    if (bq >= BS * QN) return;
    const int q = bq % QN;
    const int b = bq / QN;

    const float* vbase = value + (size_t)(b * NH + h) * TOTAL * DH;
    const float* op    = off   + ((size_t)(b * QN + q) * NH + h) * (NL * NPNT * 2);
    const float* ap    = attn  + ((size_t)(b * QN + q) * NH + h) * 16;
    const float* rp    = refp  + (size_t)(b * QN + q) * NL * 2;

    float acc = 0.0f;

    #pragma unroll
    for (int l = 0; l < NL; ++l) {
        const int   Wl = Ws[l], Hl = Hs[l];
        const float Wf = (float)Wl, Hf = (float)Hl;
        const float rx = rp[l * 2 + 0];
        const float ry = rp[l * 2 + 1];
        const float* vl = vbase + (size_t)St[l] * DH;

        #pragma unroll
        for (int p = 0; p < NPNT; ++p) {
            const float ox = op[(l * NPNT + p) * 2 + 0];
            const float oy = op[(l * NPNT + p) * 2 + 1];
            const float aw = ap[l * NPNT + p];

            const float x = rx * Wf + ox - 0.5f;
            const float y = ry * Hf + oy - 0.5f;
            const float x0f = floorf(x), y0f = floorf(y);
            const float wx1 = x - x0f,  wy1 = y - y0f;
            const float wx0 = 1.0f - wx1, wy0 = 1.0f - wy1;
            const int x0 = (int)x0f, y0 = (int)y0f;
            const int x1 = x0 + 1,  y1 = y0 + 1;

            const bool vx0 = (x0 >= 0) & (x0 < Wl);
            const bool vx1 = (x1 >= 0) & (x1 < Wl);
            const bool vy0 = (y0 >= 0) & (y0 < Hl);
            const bool vy1 = (y1 >= 0) & (y1 < Hl);

            const int cx0 = min(max(x0, 0), Wl - 1);
            const int cx1 = min(max(x1, 0), Wl - 1);
            const int cy0 = min(max(y0, 0), Hl - 1);
            const int cy1 = min(max(y1, 0), Hl - 1);

            const float g00 = vl[((size_t)(cy0 * Wl + cx0)) * DH + lane];
            const float g10 = vl[((size_t)(cy0 * Wl + cx1)) * DH + lane];
            const float g01 = vl[((size_t)(cy1 * Wl + cx0)) * DH + lane];
            const float g11 = vl[((size_t)(cy1 * Wl + cx1)) * DH + lane];

            const float w00 = (vx0 & vy0) ? wx0 * wy0 : 0.0f;
            const float w10 = (vx1 & vy0) ? wx1 * wy0 : 0.0f;
            const float w01 = (vx0 & vy1) ? wx0 * wy1 : 0.0f;
            const float w11 = (vx1 & vy1) ? wx1 * wy1 : 0.0f;

            acc += aw * (g00 * w00 + g10 * w10 + g01 * w01 + g11 * w11);
        }
    }

    mid[((size_t)(b * QN + q) * NH + h) * DH + lane] = acc;
}

// =====================================================================
// launch
// =====================================================================
extern "C" void kernel_launch(void* const* d_in, const int* in_sizes, int n_in,
                              void* d_out, int out_size, void* d_ws, size_t ws_size,
                              hipStream_t stream)
{
    const float* query  = (const float*)d_in[0];
    const float* refp   = (const float*)d_in[1];
    const float* inflat = (const float*)d_in[2];
    const float* W_off  = (const float*)d_in[3];
    const float* b_off  = (const float*)d_in[4];
    const float* W_attn = (const float*)d_in[5];
    const float* b_attn = (const float*)d_in[6];
    const float* W_v    = (const float*)d_in[7];
    const float* b_v    = (const float*)d_in[8];
    const float* W_o    = (const float*)d_in[9];
    const float* b_o    = (const float*)d_in[10];
    float* out = (float*)d_out;

    // workspace carve-up (floats)
    float* ws    = (float*)d_ws;
    float* value = ws;                                         // 40,851,456
    float* offr  = value + (size_t)BS * NH * TOTAL * DH;       //  1,843,200
    float* attnr = offr  + (size_t)BS * QN * 256;              //    921,600
    float* mid   = attnr + (size_t)BS * QN * (NH * NL * NPNT); //  1,843,200

    const dim3 blk(256);
    const int M_val = BS * TOTAL;   // 159576
    const int M_q   = BS * QN;      // 7200

    // 1) value projection (fused transpose to [bs,NH,TOTAL,DH])
    gemm256_wmma<256, 1><<<dim3((M_val + 15) / 16), blk, 0, stream>>>(
        inflat, W_v, b_v, value, M_val);

    // 2) sampling-offset projection
    gemm256_wmma<256, 0><<<dim3((M_q + 15) / 16), blk, 0, stream>>>(
        query, W_off, b_off, offr, M_q);

    // 3) attention-logit projection (N=128)
    gemm256_wmma<128, 0><<<dim3((M_q + 15) / 16), blk, 0, stream>>>(
        query, W_attn, b_attn, attnr, M_q);

    // 4) softmax over 16 logits per (b,q,h)
    const int rows = BS * QN * NH;  // 57600
    softmax16_kernel<<<(rows + 255) / 256, 256, 0, stream>>>(attnr, rows);

    // 5) deformable sampling + weighted reduce: one wave per (b,q,h)
    const int nwaves = BS * QN * NH;  // 57600 waves
    ms_deform_sample_kernel<<<(nwaves * 32 + 255) / 256, 256, 0, stream>>>(
        value, refp, offr, attnr, mid);

    // 6) output projection
    gemm256_wmma<256, 0><<<dim3((M_q + 15) / 16), blk, 0, stream>>>(
        mid, W_o, b_o, out, M_q);
}